// RankLoss_79182017069844
// MI455X (gfx1250) — compile-verified
//
#include <hip/hip_runtime.h>
#include <stdint.h>
#include <stddef.h>

#define NMAX    128   // B*K in the harness (128)
#define GRID_A  256
#define BLOCK_A 256

// ---------------------------------------------------------------------------
// CDNA5 async copy: global -> LDS, 16 bytes per active lane, ASYNCcnt-tracked.
//   lds_off : LDS byte address (low 32 bits of a generic shared pointer)
//   gbase   : 64-bit global base (SADDR pair), voff : per-lane byte offset (GVS)
// ---------------------------------------------------------------------------
__device__ __forceinline__ void async_load16(uint32_t lds_off, const void* gbase,
                                             uint32_t voff) {
  asm volatile("global_load_async_to_lds_b128 %0, %1, %2"
               :: "v"(lds_off), "v"(voff), "s"((uint64_t)(uintptr_t)gbase)
               : "memory");
}
__device__ __forceinline__ void wait_async0() {
  asm volatile("s_wait_asynccnt 0" ::: "memory");
}

__device__ __forceinline__ uint32_t lds_addr(const void* shared_ptr) {
  // generic pointer to LDS: low 32 bits are the LDS byte offset
  return (uint32_t)(uintptr_t)shared_ptr;
}

// ---------------------------------------------------------------------------
// Kernel A: masked suppression sweep over all (n, a) pairs.
// Per-block partial loss (double) + mask count (int), fixed block assignment.
// ---------------------------------------------------------------------------
__global__ void __launch_bounds__(BLOCK_A)
rank_partial_kernel(const float* __restrict__ cls,        // (B, A, 2)
                    const float* __restrict__ anchors,    // (A, 4)
                    const float* __restrict__ patch_img,  // (N, 4)
                    const float* __restrict__ patch_score,// (N, 2)
                    const int*   __restrict__ bbox_id,    // (N)
                    const int*   __restrict__ label,      // (N)
                    int A, int B, int K, int N,
                    double* __restrict__ lossPart, int* __restrict__ countPart) {
  __shared__ __align__(16) float s_patch[NMAX * 4];
  __shared__ __align__(16) float s_score[NMAX * 2];
  __shared__ __align__(16) int   s_bid[NMAX];
  __shared__ __align__(16) int   s_lab[NMAX];
  __shared__ double s_red[BLOCK_A];
  __shared__ int    s_cnt[BLOCK_A];

  const int tid = threadIdx.x;

  // ---- Stage the small per-n tables into LDS with async-tensor path ----
  const int cPatch = (N * 16) / 16;  // 16B chunks
  const int cScore = (N * 8)  / 16;
  const int cInt   = (N * 4)  / 16;
  for (int c = tid; c < cPatch; c += blockDim.x)
    async_load16(lds_addr(s_patch) + (uint32_t)c * 16u, patch_img,  (uint32_t)c * 16u);
  for (int c = tid; c < cScore; c += blockDim.x)
    async_load16(lds_addr(s_score) + (uint32_t)c * 16u, patch_score,(uint32_t)c * 16u);
  for (int c = tid; c < cInt; c += blockDim.x)
    async_load16(lds_addr(s_bid)   + (uint32_t)c * 16u, bbox_id,    (uint32_t)c * 16u);
  for (int c = tid; c < cInt; c += blockDim.x)
    async_load16(lds_addr(s_lab)   + (uint32_t)c * 16u, label,      (uint32_t)c * 16u);
  wait_async0();
  __syncthreads();

  const int stride = gridDim.x * blockDim.x;
  double acc = 0.0;
  int    cnt = 0;

  const float4* anch4 = (const float4*)anchors;
  for (int a = blockIdx.x * blockDim.x + tid; a < A; a += stride) {
    const float4 an = anch4[a];
    const float area_a = (an.z - an.x) * (an.w - an.y);

    for (int j = 0; j < B; ++j) {
      // C == 2 (harness); clip, and precompute log(1-c) per batch row.
      const float2 cv = *(const float2*)(cls + ((size_t)j * A + a) * 2);
      float c0 = fminf(fmaxf(cv.x, 1e-5f), 1.0f - 1e-5f);
      float c1 = fminf(fmaxf(cv.y, 1e-5f), 1.0f - 1e-5f);
      float l0 = logf(1.0f - c0);
      float l1 = logf(1.0f - c1);

      const int base = j * K;
      for (int k = 0; k < K; ++k) {
        const int n = base + k;
        const float px1 = s_patch[n * 4 + 0];
        const float py1 = s_patch[n * 4 + 1];
        const float px2 = s_patch[n * 4 + 2];
        const float py2 = s_patch[n * 4 + 3];
        float iw = fmaxf(fminf(an.z, px2) - fmaxf(an.x, px1), 0.0f);
        float ih = fmaxf(fminf(an.w, py2) - fmaxf(an.y, py1), 0.0f);
        float inter  = iw * ih;
        float area_b = (px2 - px1) * (py2 - py1);
        float ua  = fmaxf(area_a + area_b - inter, 1e-8f);
        float iou = inter / ua;
        if (iou >= 0.6f && a != s_bid[n]) {
          cnt++;
          const float ps0 = s_score[n * 2 + 0];
          const float ps1 = s_score[n * 2 + 1];
          float t;
          if (s_lab[n] == 0) {
            // match reference association: (ps1 - (1 - c)) + 0.05
            t = (fmaxf((ps1 - (1.0f - c0)) + 0.05f, 0.0f) - l0)
              + (fmaxf((ps1 - (1.0f - c1)) + 0.05f, 0.0f) - l1);
          } else {
            t = fmaxf((ps0 - c0) + 0.05f, 0.0f)
              + fmaxf((ps0 - c1) + 0.05f, 0.0f);
          }
          acc += (double)t;
        }
      }
    }
  }

  s_red[tid] = acc;
  s_cnt[tid] = cnt;
  __syncthreads();
  for (int s = BLOCK_A / 2; s > 0; s >>= 1) {
    if (tid < s) { s_red[tid] += s_red[tid + s]; s_cnt[tid] += s_cnt[tid + s]; }
    __syncthreads();
  }
  if (tid == 0) {
    lossPart[blockIdx.x]  = s_red[0];
    countPart[blockIdx.x] = s_cnt[0];
  }
}

// ---------------------------------------------------------------------------
// Kernel B: regression loss, one block (one wave) per batch.
// Only anchors named by bbox_id[j] with label==1 can be positive; dedup by
// first occurrence (== scatter-max semantics of the reference).
// ---------------------------------------------------------------------------
__global__ void __launch_bounds__(32)
reg_kernel(const float* __restrict__ anchors,      // (A,4)
           const float* __restrict__ annotations,  // (B,G,5)
           const float* __restrict__ regressions,  // (B,A,4)
           const int*   __restrict__ bbox_id,      // (B,K)
           const int*   __restrict__ label,        // (B*K)
           int A, int K, int G, float* __restrict__ regOut) {
  const int j   = blockIdx.x;
  const int tid = threadIdx.x;
  __shared__ int   s_id[32];
  __shared__ int   s_pos[32];
  __shared__ float s_sl[32];
  __shared__ int   s_own[32];

  if (tid < K) {
    s_id[tid]  = bbox_id[j * K + tid];
    s_pos[tid] = (label[j * K + tid] == 1) ? 1 : 0;
  } else {
    s_id[tid]  = -1;
    s_pos[tid] = 0;
  }
  __syncthreads();

  float sl  = 0.0f;
  int   own = 0;
  if (tid < K && s_pos[tid]) {
    const int a = s_id[tid];
    own = 1;
    for (int kk = 0; kk < tid; ++kk)
      if (s_pos[kk] && s_id[kk] == a) { own = 0; break; }
    if (own) {
      const float4 an = ((const float4*)anchors)[a];
      const float aw  = an.z - an.x;
      const float ah  = an.w - an.y;
      const float acx = an.x + 0.5f * aw;
      const float acy = an.y + 0.5f * ah;
      const float area_a = aw * ah;

      float best = -1.0f;  // first strict max wins, like jnp.argmax
      int   bi   = 0;
      for (int g = 0; g < G; ++g) {
        const float* bb = annotations + ((size_t)j * G + g) * 5;
        const float bx1 = bb[0], by1 = bb[1], bx2 = bb[2], by2 = bb[3];
        float iw = fmaxf(fminf(an.z, bx2) - fmaxf(an.x, bx1), 0.0f);
        float ih = fmaxf(fminf(an.w, by2) - fmaxf(an.y, by1), 0.0f);
        float inter = iw * ih;
        float ab    = (bx2 - bx1) * (by2 - by1);
        float ua    = fmaxf(area_a + ab - inter, 1e-8f);
        float iou   = inter / ua;
        if (iou > best) { best = iou; bi = g; }
      }
      const float* bb = annotations + ((size_t)j * G + bi) * 5;
      float gw  = bb[2] - bb[0];
      float gh  = bb[3] - bb[1];
      float gcx = bb[0] + 0.5f * gw;   // centers BEFORE clamping (reference order)
      float gcy = bb[1] + 0.5f * gh;
      gw = fmaxf(gw, 1.0f);
      gh = fmaxf(gh, 1.0f);
      const float t0 = ((gcx - acx) / aw) / 0.1f;
      const float t1 = ((gcy - acy) / ah) / 0.1f;
      const float t2 = logf(gw / aw) / 0.2f;
      const float t3 = logf(gh / ah) / 0.2f;
      const float* rg = regressions + ((size_t)j * A + a) * 4;
      const float d0 = fabsf(t0 - rg[0]);
      const float d1 = fabsf(t1 - rg[1]);
      const float d2 = fabsf(t2 - rg[2]);
      const float d3 = fabsf(t3 - rg[3]);
      const float th = 1.0f / 9.0f;
      const float c  = 0.5f / 9.0f;
      sl = (d0 <= th ? 4.5f * d0 * d0 : d0 - c)
         + (d1 <= th ? 4.5f * d1 * d1 : d1 - c)
         + (d2 <= th ? 4.5f * d2 * d2 : d2 - c)
         + (d3 <= th ? 4.5f * d3 * d3 : d3 - c);
    }
  }
  s_sl[tid]  = sl;
  s_own[tid] = own;
  __syncthreads();
  if (tid == 0) {
    float s  = 0.0f;
    int   np = 0;
    for (int i = 0; i < 32; ++i) { s += s_sl[i]; np += s_own[i]; }
    regOut[j] = (np > 0) ? s / ((float)np * 4.0f) : 0.0f;
  }
}

// ---------------------------------------------------------------------------
// Kernel C: deterministic finalize. Sums block partials + sup_main, divides
// by rank_count; averages reg losses.
// ---------------------------------------------------------------------------
__global__ void __launch_bounds__(256)
finalize_kernel(const double* __restrict__ lossPart, const int* __restrict__ countPart,
                int nPart, const float* __restrict__ regOut,
                const float* __restrict__ cls, const float* __restrict__ patch_score,
                const int* __restrict__ bbox_id, const int* __restrict__ label,
                int A, int B, int K, int N, float* __restrict__ out) {
  __shared__ double s_red[256];
  __shared__ int    s_cnt[256];
  const int tid = threadIdx.x;

  double acc = (tid < nPart) ? lossPart[tid] : 0.0;
  int    cnt = (tid < nPart) ? countPart[tid] : 0;

  if (tid < N) {
    const int n    = tid;
    const int j    = n / K;
    const int idxn = bbox_id[n];
    const float ps0 = patch_score[n * 2 + 0];
    const float ps1 = patch_score[n * 2 + 1];
    const int   isz = (label[n] == 0);
    const float* cp = cls + ((size_t)j * A + idxn) * 2;
    float sm = 0.0f;
    for (int c = 0; c < 2; ++c) {
      const float v = fminf(fmaxf(cp[c], 1e-5f), 1.0f - 1e-5f);
      if (isz)
        sm += fmaxf((ps1 - (1.0f - v)) + 0.05f, 0.0f) - 0.5f * logf(1.0f - v);
      else
        sm += fmaxf((ps0 - v) + 0.05f, 0.0f);
    }
    acc += (double)sm;
  }

  s_red[tid] = acc;
  s_cnt[tid] = cnt;
  __syncthreads();
  for (int s = 128; s > 0; s >>= 1) {
    if (tid < s) { s_red[tid] += s_red[tid + s]; s_cnt[tid] += s_cnt[tid + s]; }
    __syncthreads();
  }
  if (tid == 0) {
    const double rank = s_red[0] / ((double)N + (double)s_cnt[0]);
    out[0] = (float)rank;
    float rs = 0.0f;
    for (int j = 0; j < B; ++j) rs += regOut[j];
    out[1] = rs / (float)B;
  }
}

// ---------------------------------------------------------------------------
extern "C" void kernel_launch(void* const* d_in, const int* in_sizes, int n_in,
                              void* d_out, int out_size, void* d_ws, size_t ws_size,
                              hipStream_t stream) {
  const float* patch_score = (const float*)d_in[0];
  const float* cls         = (const float*)d_in[1];
  const int*   bbox_id     = (const int*)  d_in[2];
  // d_in[3] = batch_num (device scalar) — B derived from sizes instead
  const float* regressions = (const float*)d_in[4];
  const float* anchors     = (const float*)d_in[5];
  const float* annotations = (const float*)d_in[6];
  const int*   label       = (const int*)  d_in[7];
  const float* patch_img   = (const float*)d_in[8];

  const int A = in_sizes[5] / 4;             // anchors: (1, A, 4)
  const int N = in_sizes[0] / 2;             // patch_score: (N, 2)
  const int B = in_sizes[1] / (A * 2);       // classifications: (B, A, 2)
  const int K = N / B;                       // bbox_id: (B, K)
  const int G = in_sizes[6] / (5 * B);       // annotations: (B, G, 5)

  double* lossPart  = (double*)d_ws;
  int*    countPart = (int*)((char*)d_ws + GRID_A * sizeof(double));
  float*  regOut    = (float*)((char*)d_ws + GRID_A * (sizeof(double) + sizeof(int)));

  rank_partial_kernel<<<GRID_A, BLOCK_A, 0, stream>>>(
      cls, anchors, patch_img, patch_score, bbox_id, label,
      A, B, K, N, lossPart, countPart);

  reg_kernel<<<B, 32, 0, stream>>>(
      anchors, annotations, regressions, bbox_id, label, A, K, G, regOut);

  finalize_kernel<<<1, 256, 0, stream>>>(
      lossPart, countPart, GRID_A, regOut, cls, patch_score, bbox_id, label,
      A, B, K, N, (float*)d_out);
}